// GatedDeltaNet_74517682586448
// MI455X (gfx1250) — compile-verified
//
#include <hip/hip_runtime.h>
#include <hip/hip_bf16.h>
#include <math.h>

// Gated Delta Rule (chunked), fp32 WMMA (V_WMMA_F32_16X16X4_F32), gfx1250.
// grid = B*H*VSPL workgroups; each owns (b, h, Dv-block) and scans 64 chunks
// sequentially with the [128 x 32] recurrent state resident in LDS.
// UT transform done as blocked (I-A)^-1 forward substitution with WMMA.
// wave id forced scalar (readfirstlane) and per-tile conditions hoisted out of
// K loops so tile control flow is SALU branches, not per-iteration EXEC masks.

typedef __attribute__((ext_vector_type(2))) float f32x2;
typedef __attribute__((ext_vector_type(8))) float f32x8;

#define BB   2
#define SS   4096
#define HH   16
#define DK   128
#define DV   128
#define CC   64
#define NCH  (SS / CC)
#define VSPL 4
#define DVB  (DV / VSPL)   // 32 Dv columns per workgroup

// LDS row strides (padded to dodge bank conflicts)
#define LQ 132   // sq / sk / skc rows of 128
#define LA 68    // sattn rows of 64
#define LV 36    // sv-like rows of 32

#define NTHREADS 256

static __device__ __forceinline__ f32x8 wmma4(f32x2 a, f32x2 b, f32x8 c) {
  // D = A(16x4) * B(4x16) + C(16x16), fp32
  return __builtin_amdgcn_wmma_f32_16x16x4_f32(false, a, false, b,
                                               (short)0, c, false, false);
}

// 16x16x16 matmul: acc += A(16x16, row-major, stride lda) @ B(16x16, stride ldb)
static __device__ __forceinline__ f32x8
mm16(const float* __restrict__ A, int lda, const float* __restrict__ Bm, int ldb,
     f32x8 acc, int lm, int kp) {
  #pragma unroll
  for (int kk = 0; kk < 16; kk += 4) {
    f32x2 a, b;
    a.x = A[lm * lda + kk + kp];
    a.y = A[lm * lda + kk + kp + 1];
    b.x = Bm[(kk + kp) * ldb + lm];
    b.y = Bm[(kk + kp + 1) * ldb + lm];
    acc = wmma4(a, b, acc);
  }
  return acc;
}

__global__ __launch_bounds__(NTHREADS)
void gdn_chunk_scan(const float* __restrict__ q_g, const float* __restrict__ k_g,
                    const float* __restrict__ v_g, const float* __restrict__ g_g,
                    const float* __restrict__ b_g, float* __restrict__ out) {
  extern __shared__ float smem[];
  float* sq     = smem;                 // [64][LQ] l2norm(q)/sqrt(Dk)
  float* sk     = sq     + 64 * LQ;     // [64][LQ] l2norm(k)
  float* skc    = sk     + 64 * LQ;     // [64][LQ] k_cum
  float* sattn  = skc    + 64 * LQ;     // [64][LA] A -> T=(I-A)^-1 -> attn_l
  float* sv     = sattn  + 64 * LA;     // [64][LV] raw v (this Dv block)
  float* svn    = sv     + 64 * LV;     // [64][LV] v_new
  float* sstate = svn    + 64 * LV;     // [128][LV] recurrent state
  float* gcs    = sstate + 128 * LV;    // [64] cumsum(g)
  float* sbeta  = gcs    + 64;          // [64]
  float* qg     = sbeta  + 64;          // [64] raw g, then exp(gc)
  float* bexp   = qg     + 64;          // [64] beta*exp(gc)
  float* kdec   = bexp   + 64;          // [64] exp(g_last - gc)
  float* red    = kdec   + 64;          // [64][8] sumsq partials (q | k)
  float* pscr   = red    + 64 * 8;      // [4][256] per-wave P scratch (UT)

  const int tid  = threadIdx.x;
  const int lane = tid & 31;
  // wave id is uniform per wave; force it into an SGPR so all tile-assignment
  // control flow lowers to scalar branches (no EXEC-mask churn around WMMA).
  const int wave = __builtin_amdgcn_readfirstlane(tid >> 5);
  const int bid  = blockIdx.x;
  const int b    = bid / (HH * VSPL);
  const int h    = (bid / VSPL) % HH;
  const int v0   = (bid % VSPL) * DVB;

  const int lm = lane & 15;            // M/N index within fragment
  const int hi = lane >> 4;            // lane-half (selects K pair / M+8)
  const int kp = hi << 1;              // K offset within 4-wide K slab
  const int mh = hi << 3;              // +8 row offset for C/D high half

  for (int i = tid; i < 128 * LV; i += NTHREADS) sstate[i] = 0.f;
  __syncthreads();

  const int row = tid >> 2;            // 64 rows, 4 threads per row
  const int sub = tid & 3;

  for (int n = 0; n < NCH; ++n) {
    const int c0 = n * CC;

    // ---------------- load + l2norm + cumsum(g) ----------------
    {
      size_t baseqk = (((size_t)b * SS + c0 + row) * HH + h) * (size_t)DK + sub * 32;
      float sumq = 0.f, sumk = 0.f;
      #pragma unroll 8
      for (int i = 0; i < 32; ++i) {
        float vq = q_g[baseqk + i];
        float vk = k_g[baseqk + i];
        sq[row * LQ + sub * 32 + i] = vq;
        sk[row * LQ + sub * 32 + i] = vk;
        sumq += vq * vq;
        sumk += vk * vk;
      }
      red[row * 8 + sub]     = sumq;
      red[row * 8 + 4 + sub] = sumk;
      size_t basev = (((size_t)b * SS + c0 + row) * HH + h) * (size_t)DV + v0 + sub * 8;
      #pragma unroll
      for (int i = 0; i < 8; ++i)
        sv[row * LV + sub * 8 + i] = v_g[basev + i];
      if (tid < 64) {
        size_t gi = ((size_t)b * SS + c0 + tid) * HH + h;
        qg[tid]    = g_g[gi];          // raw g (temp)
        sbeta[tid] = b_g[gi];
      }
    }
    __syncthreads();
    if (tid == 0) {                    // per-chunk cumulative log-decay
      float run = 0.f;
      for (int c = 0; c < 64; ++c) { run += qg[c]; gcs[c] = run; }
    }
    __syncthreads();
    {
      float sumq = red[row*8+0] + red[row*8+1] + red[row*8+2] + red[row*8+3];
      float sumk = red[row*8+4] + red[row*8+5] + red[row*8+6] + red[row*8+7];
      float rq = rsqrtf(sumq + 1e-6f) * 0.08838834764831845f;  // 1/sqrt(128) folded
      float rk = rsqrtf(sumk + 1e-6f);
      #pragma unroll 8
      for (int i = 0; i < 32; ++i) {
        sq[row * LQ + sub * 32 + i] *= rq;
        sk[row * LQ + sub * 32 + i] *= rk;
      }
      if (tid < 64) {
        float e = __expf(gcs[tid]);
        qg[tid]   = e;                              // exp(gc)
        bexp[tid] = sbeta[tid] * e;                 // beta*exp(gc)
        kdec[tid] = __expf(gcs[63] - gcs[tid]);     // exp(g_last - gc)
      }
    }
    __syncthreads();

    // ------- step 1: A = -(k_beta @ k^T) * decay, strict lower -------
    // wave -> row-tile ti = wave>>1, tj pair {2*(wave&1), +1}; A frag shared.
    {
      const int ti  = wave >> 1;
      const int tj0 = (wave & 1) * 2;
      const int tj1 = tj0 + 1;
      const int mrow = ti * 16 + lm;
      const int n0 = tj0 * 16 + lm, n1 = tj1 * 16 + lm;
      const bool v0t = (ti >= tj0), v1t = (ti >= tj1);
      if (!v0t) {
        #pragma unroll
        for (int r = 0; r < 8; ++r) sattn[(ti*16 + r + mh) * LA + n0] = 0.f;
      }
      if (!v1t) {
        #pragma unroll
        for (int r = 0; r < 8; ++r) sattn[(ti*16 + r + mh) * LA + n1] = 0.f;
      }
      if (v0t) {
        f32x8 acc0 = {}, acc1 = {};
        float bm = sbeta[mrow];
        if (v1t) {              // both tiles valid: shared A, 2 accumulators
          #pragma unroll 8
          for (int kk = 0; kk < DK; kk += 4) {
            f32x2 a, b0, b1;
            a.x = sk[mrow * LQ + kk + kp]     * bm;
            a.y = sk[mrow * LQ + kk + kp + 1] * bm;
            b0.x = sk[n0 * LQ + kk + kp];
            b0.y = sk[n0 * LQ + kk + kp + 1];
            b1.x = sk[n1 * LQ + kk + kp];
            b1.y = sk[n1 * LQ + kk + kp + 1];
            acc0 = wmma4(a, b0, acc0);
            acc1 = wmma4(a, b1, acc1);
          }
        } else {                // diagonal-boundary wave: single tile
          #pragma unroll 8
          for (int kk = 0; kk < DK; kk += 4) {
            f32x2 a, b0;
            a.x = sk[mrow * LQ + kk + kp]     * bm;
            a.y = sk[mrow * LQ + kk + kp + 1] * bm;
            b0.x = sk[n0 * LQ + kk + kp];
            b0.y = sk[n0 * LQ + kk + kp + 1];
            acc0 = wmma4(a, b0, acc0);
          }
        }
        #pragma unroll
        for (int r = 0; r < 8; ++r) {
          int m = ti * 16 + r + mh;
          float o = (m > n0) ? -acc0[r] * __expf(gcs[m] - gcs[n0]) : 0.f;
          sattn[m * LA + n0] = o;
        }
        if (v1t) {
          #pragma unroll
          for (int r = 0; r < 8; ++r) {
            int m = ti * 16 + r + mh;
            float o = (m > n1) ? -acc1[r] * __expf(gcs[m] - gcs[n1]) : 0.f;
            sattn[m * LA + n1] = o;
          }
        }
      }
    }
    __syncthreads();

    // ------- step 2: blocked UT transform: sattn <- (I-A)^-1 - I -------
    // phase a: in-place row-update on 4 diagonal 16x16 blocks (X_dd)
    for (int ii = 1; ii < 16; ++ii) {
      float s = 0.f;
      if (tid < 64) {
        int d = tid >> 4, jj = tid & 15;
        const float* base = sattn + (d * 16) * LA + d * 16;
        for (int c = 0; c < ii; ++c)
          s += base[ii * LA + c] * base[c * LA + jj];
      }
      __syncthreads();
      if (tid < 64) {
        int d = tid >> 4, jj = tid & 15;
        sattn[(d * 16 + ii) * LA + d * 16 + jj] += s;
      }
      __syncthreads();
    }
    // phase b: off-diagonal blocks, block-row i = 1..3, wave j<i computes T_ij
    for (int i = 1; i < 4; ++i) {
      f32x8 tij = {};
      const bool active = (wave < i);
      if (active) {
        int j = wave;
        // P = A_ij + A_ij @ X_jj + sum_{c=j+1}^{i-1} A_ic @ T_cj
        f32x8 p;
        #pragma unroll
        for (int r = 0; r < 8; ++r)
          p[r] = sattn[(i * 16 + r + mh) * LA + j * 16 + lm];
        p = mm16(sattn + (i*16)*LA + j*16, LA, sattn + (j*16)*LA + j*16, LA, p, lm, kp);
        for (int c = j + 1; c < i; ++c)
          p = mm16(sattn + (i*16)*LA + c*16, LA, sattn + (c*16)*LA + j*16, LA, p, lm, kp);
        // stage P in per-wave scratch so it can feed the B operand
        float* ps = pscr + wave * 256;
        #pragma unroll
        for (int r = 0; r < 8; ++r) ps[(r + mh) * 16 + lm] = p[r];
        // T_ij = P + X_ii @ P   (same-wave LDS RAW: in-order + dscnt wait)
        tij = mm16(sattn + (i*16)*LA + i*16, LA, ps, 16, p, lm, kp);
      }
      __syncthreads();
      if (active) {
        int j = wave;
        #pragma unroll
        for (int r = 0; r < 8; ++r)
          sattn[(i * 16 + r + mh) * LA + j * 16 + lm] = tij[r];
      }
      __syncthreads();
    }
    if (tid < 64) sattn[tid * LA + tid] += 1.0f;   // T = I + X
    __syncthreads();

    // ------- step 4: k_cum = T @ (k * beta * exp(gc)) -------
    // wave -> ti = wave>>1; two passes of 2 Dk tiles, A frag shared per pass.
    {
      const int ti   = wave >> 1;
      const int mrow = ti * 16 + lm;
      #pragma unroll
      for (int half = 0; half < 2; ++half) {
        int tkb = (wave & 1) * 4 + half * 2;
        int n0 = tkb * 16 + lm, n1 = (tkb + 1) * 16 + lm;
        f32x8 acc0 = {}, acc1 = {};
        #pragma unroll 8
        for (int kk = 0; kk < CC; kk += 4) {
          f32x2 a, b0, b1;
          a.x = sattn[mrow * LA + kk + kp];
          a.y = sattn[mrow * LA + kk + kp + 1];
          float e0 = bexp[kk + kp], e1 = bexp[kk + kp + 1];
          b0.x = sk[(kk + kp)     * LQ + n0] * e0;
          b0.y = sk[(kk + kp + 1) * LQ + n0] * e1;
          acc0 = wmma4(a, b0, acc0);
          b1.x = sk[(kk + kp)     * LQ + n1] * e0;
          b1.y = sk[(kk + kp + 1) * LQ + n1] * e1;
          acc1 = wmma4(a, b1, acc1);
        }
        #pragma unroll
        for (int r = 0; r < 8; ++r) {
          skc[(ti * 16 + r + mh) * LQ + n0] = acc0[r];
          skc[(ti * 16 + r + mh) * LQ + n1] = acc1[r];
        }
      }
    }
    __syncthreads();

    // ------- fused steps 3+6: v_new = T @ (v*beta) - k_cum @ state -------
    {
      const int ti = wave >> 1, tv = wave & 1;
      const int mrow = ti * 16 + lm;
      const int ncol = tv * 16 + lm;
      f32x8 acc = {};
      #pragma unroll 8
      for (int kk = 0; kk < CC; kk += 4) {          // T @ (v*beta)
        f32x2 a, bbf;
        a.x  = sattn[mrow * LA + kk + kp];
        a.y  = sattn[mrow * LA + kk + kp + 1];
        bbf.x = sv[(kk + kp)     * LV + ncol] * sbeta[kk + kp];
        bbf.y = sv[(kk + kp + 1) * LV + ncol] * sbeta[kk + kp + 1];
        acc = wmma4(a, bbf, acc);
      }
      #pragma unroll 8
      for (int kk = 0; kk < DK; kk += 4) {          // - k_cum @ state
        f32x2 a, bbf;
        a.x  = -skc[mrow * LQ + kk + kp];
        a.y  = -skc[mrow * LQ + kk + kp + 1];
        bbf.x = sstate[(kk + kp)     * LV + ncol];
        bbf.y = sstate[(kk + kp + 1) * LV + ncol];
        acc = wmma4(a, bbf, acc);
      }
      #pragma unroll
      for (int r = 0; r < 8; ++r)
        svn[(ti * 16 + r + mh) * LV + ncol] = acc[r];
    }
    __syncthreads();

    // ------- step 5: attn_l = tril(q @ k^T) * decay  (overwrites sattn) -------
    {
      const int ti  = wave >> 1;
      const int tj0 = (wave & 1) * 2;
      const int tj1 = tj0 + 1;
      const int mrow = ti * 16 + lm;
      const int n0 = tj0 * 16 + lm, n1 = tj1 * 16 + lm;
      const bool v0t = (ti >= tj0), v1t = (ti >= tj1);
      if (!v0t) {
        #pragma unroll
        for (int r = 0; r < 8; ++r) sattn[(ti*16 + r + mh) * LA + n0] = 0.f;
      }
      if (!v1t) {
        #pragma unroll
        for (int r = 0; r < 8; ++r) sattn[(ti*16 + r + mh) * LA + n1] = 0.f;
      }
      if (v0t) {
        f32x8 acc0 = {}, acc1 = {};
        if (v1t) {              // both tiles valid: shared A, 2 accumulators
          #pragma unroll 8
          for (int kk = 0; kk < DK; kk += 4) {
            f32x2 a, b0, b1;
            a.x = sq[mrow * LQ + kk + kp];
            a.y = sq[mrow * LQ + kk + kp + 1];
            b0.x = sk[n0 * LQ + kk + kp];
            b0.y = sk[n0 * LQ + kk + kp + 1];
            b1.x = sk[n1 * LQ + kk + kp];
            b1.y = sk[n1 * LQ + kk + kp + 1];
            acc0 = wmma4(a, b0, acc0);
            acc1 = wmma4(a, b1, acc1);
          }
        } else {                // diagonal-boundary wave: single tile
          #pragma unroll 8
          for (int kk = 0; kk < DK; kk += 4) {
            f32x2 a, b0;
            a.x = sq[mrow * LQ + kk + kp];
            a.y = sq[mrow * LQ + kk + kp + 1];
            b0.x = sk[n0 * LQ + kk + kp];
            b0.y = sk[n0 * LQ + kk + kp + 1];
            acc0 = wmma4(a, b0, acc0);
          }
        }
        #pragma unroll
        for (int r = 0; r < 8; ++r) {
          int m = ti * 16 + r + mh;
          float o = (m >= n0) ? acc0[r] * __expf(gcs[m] - gcs[n0]) : 0.f;
          sattn[m * LA + n0] = o;
        }
        if (v1t) {
          #pragma unroll
          for (int r = 0; r < 8; ++r) {
            int m = ti * 16 + r + mh;
            float o = (m >= n1) ? acc1[r] * __expf(gcs[m] - gcs[n1]) : 0.f;
            sattn[m * LA + n1] = o;
          }
        }
      }
    }
    __syncthreads();

    // ------- step 7: out = (q*exp(gc)) @ state + attn_l @ v_new -------
    {
      const int ti = wave >> 1, tv = wave & 1;
      const int mrow = ti * 16 + lm;
      const int ncol = tv * 16 + lm;
      f32x8 acc = {};
      float qm = qg[mrow];
      #pragma unroll 8
      for (int kk = 0; kk < DK; kk += 4) {            // inter-chunk part
        f32x2 a, bbf;
        a.x  = sq[mrow * LQ + kk + kp]     * qm;
        a.y  = sq[mrow * LQ + kk + kp + 1] * qm;
        bbf.x = sstate[(kk + kp)     * LV + ncol];
        bbf.y = sstate[(kk + kp + 1) * LV + ncol];
        acc = wmma4(a, bbf, acc);
      }
      #pragma unroll 8
      for (int kk = 0; kk < CC; kk += 4) {            // intra-chunk part
        f32x2 a, bbf;
        a.x  = sattn[mrow * LA + kk + kp];
        a.y  = sattn[mrow * LA + kk + kp + 1];
        bbf.x = svn[(kk + kp)     * LV + ncol];
        bbf.y = svn[(kk + kp + 1) * LV + ncol];
        acc = wmma4(a, bbf, acc);
      }
      #pragma unroll
      for (int r = 0; r < 8; ++r) {
        int m = ti * 16 + r + mh;
        out[(((size_t)b * SS + c0 + m) * HH + h) * (size_t)DV + v0 + ncol] = acc[r];
      }
    }
    __syncthreads();

    // ------- step 8: state = exp(g_last)*state + k_dec^T @ v_new -------
    // wave -> Dk tile tk = wave; both Dv tiles share the A fragment.
    {
      float eglast = __expf(gcs[63]);
      const int tk = wave;
      const int drow = tk * 16 + lm;
      f32x8 a0, a1;
      #pragma unroll
      for (int r = 0; r < 8; ++r) {
        a0[r] = sstate[(tk * 16 + r + mh) * LV + lm]      * eglast;
        a1[r] = sstate[(tk * 16 + r + mh) * LV + 16 + lm] * eglast;
      }
      #pragma unroll 8
      for (int kk = 0; kk < CC; kk += 4) {
        f32x2 a, b0, b1;
        a.x = sk[(kk + kp)     * LQ + drow] * kdec[kk + kp];
        a.y = sk[(kk + kp + 1) * LQ + drow] * kdec[kk + kp + 1];
        b0.x = svn[(kk + kp)     * LV + lm];
        b0.y = svn[(kk + kp + 1) * LV + lm];
        b1.x = svn[(kk + kp)     * LV + 16 + lm];
        b1.y = svn[(kk + kp + 1) * LV + 16 + lm];
        a0 = wmma4(a, b0, a0);
        a1 = wmma4(a, b1, a1);
      }
      #pragma unroll
      for (int r = 0; r < 8; ++r) {
        sstate[(tk * 16 + r + mh) * LV + lm]      = a0[r];
        sstate[(tk * 16 + r + mh) * LV + 16 + lm] = a1[r];
      }
    }
    __syncthreads();   // protect sk/sv/state before next chunk's loads
  }
}

extern "C" void kernel_launch(void* const* d_in, const int* in_sizes, int n_in,
                              void* d_out, int out_size, void* d_ws, size_t ws_size,
                              hipStream_t stream) {
  (void)in_sizes; (void)n_in; (void)out_size; (void)d_ws; (void)ws_size;
  const float* q  = (const float*)d_in[0];
  const float* k  = (const float*)d_in[1];
  const float* v  = (const float*)d_in[2];
  const float* g  = (const float*)d_in[3];
  const float* bt = (const float*)d_in[4];
  float* out = (float*)d_out;

  constexpr size_t SMEM =
      (size_t)(3 * 64 * LQ + 64 * LA + 2 * 64 * LV + 128 * LV + 5 * 64 + 64 * 8
               + 4 * 256) * sizeof(float);   // ~159 KB dynamic LDS (WGP: 320 KB)

  static_assert(SMEM < 320u * 1024u, "LDS budget");
  (void)hipFuncSetAttribute((const void*)gdn_chunk_scan,
                            hipFuncAttributeMaxDynamicSharedMemorySize,
                            (int)SMEM);

  dim3 grid(BB * HH * VSPL);   // 128 workgroups
  dim3 block(NTHREADS);        // 8 wave32s
  gdn_chunk_scan<<<grid, block, SMEM, stream>>>(q, k, v, g, bt, out);
}